// GIN_27367531610541
// MI455X (gfx1250) — compile-verified
//
#include <hip/hip_runtime.h>

typedef __attribute__((ext_vector_type(2))) float v2f;
typedef __attribute__((ext_vector_type(8))) float v8f;

#define N_NODES   50000
#define N_EDGES   800000
#define DF        128
#define NUM_GRAPHS 256
#define LDS_STRIDE 132   // 128 + 4 pad floats -> bank-conflict-free strided reads

// ---------------------------------------------------------------------------
// utility: zero a float buffer
// ---------------------------------------------------------------------------
__global__ void zero_f32(float* __restrict__ p, int n) {
    int i = blockIdx.x * 256 + threadIdx.x;
    if (i < n) p[i] = 0.f;
}

// ---------------------------------------------------------------------------
// GIN aggregation: agg[dst] += h[src] (atomic scatter-add, resolves in L2)
// 32 threads per edge, 4 floats per thread (float4 gather)
// ---------------------------------------------------------------------------
__global__ void scatter_add_kernel(const int* __restrict__ src,
                                   const int* __restrict__ dst,
                                   const float* __restrict__ h,
                                   float* __restrict__ agg) {
    int gid = blockIdx.x * 256 + threadIdx.x;
    int e = gid >> 5;
    if (e >= N_EDGES) return;
    int f = (gid & 31) << 2;
    int s = src[e];
    int d = dst[e];
    const float4 v = *(const float4*)(h + (size_t)s * DF + f);
    float* p = agg + (size_t)d * DF + f;
    atomicAdd(p + 0, v.x);
    atomicAdd(p + 1, v.y);
    atomicAdd(p + 2, v.z);
    atomicAdd(p + 3, v.w);
}

// ---------------------------------------------------------------------------
// Fused GEMM:  Out = preop(A) @ B + bias,  plus per-column sum / sum-of-squares
// for training-mode BatchNorm statistics.
//   preop:  A_eff = A (+ addA)                    when bnScale == nullptr
//           A_eff = relu(A*bnScale[c] + bnShift[c]) when bnScale != nullptr
// Shapes: A[M,128], B[128,128] row-major (K x N), Out[M,128].  M % 16 == 0.
// One workgroup (8 wave32s) = one 16-row M tile; wave w owns N-tile [16w,16w+16).
// K-loop = 32 x v_wmma_f32_16x16x4_f32 from LDS.
// ---------------------------------------------------------------------------
__global__ __launch_bounds__(256) void gemm128_kernel(
    const float* __restrict__ A,
    const float* __restrict__ addA,      // nullable
    const float* __restrict__ bnScale,   // nullable
    const float* __restrict__ bnShift,   // nullable
    const float* __restrict__ B,         // [128,128] K-major rows
    const float* __restrict__ bias,      // [128]
    float* __restrict__ Out,
    float* __restrict__ statSum,         // nullable [128]
    float* __restrict__ statSq)          // nullable [128]
{
    __shared__ float As[16 * LDS_STRIDE];       //  ~8.4 KB
    __shared__ float Bt[DF * LDS_STRIDE];       // ~67.6 KB, B transposed [N][K]

    const int tid  = threadIdx.x;
    const int wave = tid >> 5;
    const int lane = tid & 31;
    const int m0   = blockIdx.x * 16;

    // ---- stage A tile with fused pre-op ----
    for (int i = tid; i < 16 * DF; i += 256) {
        int r = i >> 7, c = i & (DF - 1);
        size_t gi = (size_t)(m0 + r) * DF + c;
        float v = A[gi];
        if (addA)    v += addA[gi];
        if (bnScale) { v = v * bnScale[c] + bnShift[c]; v = v > 0.f ? v : 0.f; }
        As[r * LDS_STRIDE + c] = v;
    }
    // ---- stage B transposed (coalesced global read, one-time LDS write) ----
    for (int i = tid; i < DF * DF; i += 256) {
        int k = i >> 7, n = i & (DF - 1);
        Bt[n * LDS_STRIDE + k] = B[i];
    }
    __syncthreads();

    // ISA layouts (wave32):
    //  A 16x4 : lane -> M = lane%16 ; VGPR{0,1} hold K = 2*half, 2*half+1
    //  B 4x16 : lane -> N = lane%16 ; VGPR{0,1} hold K = 2*half, 2*half+1
    //  C 16x16: lane -> N = lane%16 ; VGPR v holds M = v + 8*half
    const int idx16 = lane & 15;
    const int half  = lane >> 4;
    const int n0    = wave * 16;

    const float* Arow = &As[idx16 * LDS_STRIDE];
    const float* Brow = &Bt[(n0 + idx16) * LDS_STRIDE];

    v8f acc = {};
    #pragma unroll
    for (int kk = 0; kk < DF; kk += 4) {
        v2f a = *(const v2f*)(Arow + kk + 2 * half);
        v2f b = *(const v2f*)(Brow + kk + 2 * half);
        acc = __builtin_amdgcn_wmma_f32_16x16x4_f32(
            /*neg_a=*/false, a, /*neg_b=*/false, b,
            /*c_mod=*/(short)0, acc, /*reuse_a=*/false, /*reuse_b=*/false);
    }

    // ---- epilogue: bias, store, per-column BN statistics ----
    const int   col  = n0 + idx16;
    const float bcol = bias ? bias[col] : 0.f;
    float s = 0.f, sq = 0.f;
    #pragma unroll
    for (int v = 0; v < 8; ++v) {
        int m = v + 8 * half;
        float val = acc[v] + bcol;
        Out[(size_t)(m0 + m) * DF + col] = val;
        s  += val;
        sq += val * val;
    }
    if (statSum) {
        atomicAdd(&statSum[col], s);   // two lanes (both halves) cover all 16 rows
        atomicAdd(&statSq[col], sq);
    }
}

// ---------------------------------------------------------------------------
// BN finalize: scale/shift from accumulated sum/sumsq (biased variance),
// then reset the accumulators for the next use.
// ---------------------------------------------------------------------------
__global__ void bn_finalize_kernel(float* __restrict__ sum, float* __restrict__ sq,
                                   const float* __restrict__ gamma,
                                   const float* __restrict__ beta,
                                   float* __restrict__ scale, float* __restrict__ shift,
                                   float invN) {
    int c = threadIdx.x;   // 128 threads
    float mean = sum[c] * invN;
    float var  = sq[c] * invN - mean * mean;
    float sc   = gamma[c] * rsqrtf(var + 1e-5f);
    scale[c] = sc;
    shift[c] = beta[c] - mean * sc;
    sum[c] = 0.f;
    sq[c]  = 0.f;
}

// ---------------------------------------------------------------------------
// h = relu(z * scale[col] + shift[col])
// ---------------------------------------------------------------------------
__global__ void bn_relu_kernel(const float* __restrict__ z,
                               const float* __restrict__ scale,
                               const float* __restrict__ shift,
                               float* __restrict__ out, int n) {
    int i = blockIdx.x * 256 + threadIdx.x;
    if (i >= n) return;
    int c = i & (DF - 1);
    float v = z[i] * scale[c] + shift[c];
    out[i] = v > 0.f ? v : 0.f;
}

// ---------------------------------------------------------------------------
// Global mean pool: atomic per-graph sums + counts, then divide.
// ---------------------------------------------------------------------------
__global__ void pool_sum_kernel(const float* __restrict__ h,
                                const int* __restrict__ batch,
                                float* __restrict__ pooled,
                                float* __restrict__ counts) {
    int gid = blockIdx.x * 256 + threadIdx.x;
    int node = gid >> 5;
    if (node >= N_NODES) return;
    int f = (gid & 31) << 2;
    int g = batch[node];
    const float4 v = *(const float4*)(h + (size_t)node * DF + f);
    float* p = pooled + (size_t)g * DF + f;
    atomicAdd(p + 0, v.x);
    atomicAdd(p + 1, v.y);
    atomicAdd(p + 2, v.z);
    atomicAdd(p + 3, v.w);
    if (f == 0) atomicAdd(&counts[g], 1.f);
}

__global__ void pool_div_kernel(float* __restrict__ pooled,
                                const float* __restrict__ counts) {
    int i = blockIdx.x * 256 + threadIdx.x;   // NUM_GRAPHS*DF = 32768
    float c = counts[i >> 7];
    c = c > 1.f ? c : 1.f;
    pooled[i] = pooled[i] / c;
}

// ---------------------------------------------------------------------------
// Orchestration
// ---------------------------------------------------------------------------
extern "C" void kernel_launch(void* const* d_in, const int* in_sizes, int n_in,
                              void* d_out, int out_size, void* d_ws, size_t ws_size,
                              hipStream_t stream) {
    const float* x     = (const float*)d_in[0];
    const int*   edge  = (const int*)d_in[1];
    const int*   src   = edge;
    const int*   dst   = edge + N_EDGES;
    const int*   batch = (const int*)d_in[2];
    // params flattened in setup_inputs() insertion order:
    // per layer: W1, b1, g1, be1, W2, b2, g2, be2  -> base 3 + 8*l ; then Wfc, bfc
    const float* Wfc = (const float*)d_in[3 + 8 * 3 + 0];
    const float* bfc = (const float*)d_in[3 + 8 * 3 + 1];

    const size_t NM = (size_t)N_NODES * DF;   // 6.4M floats
    float* ws     = (float*)d_ws;
    float* agg    = ws;                //  NM
    float* z      = agg + NM;          //  NM
    float* h      = z + NM;            //  NM
    float* sum    = h + NM;            //  128
    float* sq     = sum + DF;          //  128
    float* scaleA = sq + DF;           //  128
    float* shiftA = scaleA + DF;       //  128
    float* scaleB = shiftA + DF;       //  128
    float* shiftB = scaleB + DF;       //  128
    float* counts = shiftB + DF;       //  256
    float* pooled = counts + NUM_GRAPHS;  // 256*128

    const int gemmBlocks = N_NODES / 16;           // 3125 (exact)
    const int nmBlocks   = (int)((NM + 255) / 256);
    const float invN     = 1.0f / (float)N_NODES;

    // stats accumulators must be zero on entry (ws is poisoned)
    zero_f32<<<1, 256, 0, stream>>>(sum, 2 * DF);

    const float* hin = x;
    for (int l = 0; l < 3; ++l) {
        const float* W1  = (const float*)d_in[3 + 8 * l + 0];
        const float* b1  = (const float*)d_in[3 + 8 * l + 1];
        const float* g1  = (const float*)d_in[3 + 8 * l + 2];
        const float* be1 = (const float*)d_in[3 + 8 * l + 3];
        const float* W2  = (const float*)d_in[3 + 8 * l + 4];
        const float* b2  = (const float*)d_in[3 + 8 * l + 5];
        const float* g2  = (const float*)d_in[3 + 8 * l + 6];
        const float* be2 = (const float*)d_in[3 + 8 * l + 7];

        // agg = segment_sum(h[src], dst)
        zero_f32<<<nmBlocks, 256, 0, stream>>>(agg, (int)NM);
        scatter_add_kernel<<<(N_EDGES * 32) / 256, 256, 0, stream>>>(src, dst, hin, agg);

        // z = (h + agg) @ W1 + b1   (+ column stats)
        gemm128_kernel<<<gemmBlocks, 256, 0, stream>>>(
            hin, agg, nullptr, nullptr, W1, b1, z, sum, sq);
        bn_finalize_kernel<<<1, DF, 0, stream>>>(sum, sq, g1, be1, scaleA, shiftA, invN);

        // z2 = relu(BN1(z)) @ W2 + b2  -> reuse agg buffer   (+ column stats)
        gemm128_kernel<<<gemmBlocks, 256, 0, stream>>>(
            z, nullptr, scaleA, shiftA, W2, b2, agg, sum, sq);
        bn_finalize_kernel<<<1, DF, 0, stream>>>(sum, sq, g2, be2, scaleB, shiftB, invN);

        // h = relu(BN2(z2))
        bn_relu_kernel<<<nmBlocks, 256, 0, stream>>>(agg, scaleB, shiftB, h, (int)NM);
        hin = h;
    }

    // global mean pool
    zero_f32<<<(NUM_GRAPHS * DF + NUM_GRAPHS + 255) / 256, 256, 0, stream>>>(counts,
        NUM_GRAPHS * DF + NUM_GRAPHS);   // counts + pooled are contiguous
    pool_sum_kernel<<<(N_NODES * 32 + 255) / 256, 256, 0, stream>>>(hin, batch, pooled, counts);
    pool_div_kernel<<<(NUM_GRAPHS * DF) / 256, 256, 0, stream>>>(pooled, counts);

    // out = pooled @ Wfc + bfc    (256 rows -> 16 M-tiles)
    gemm128_kernel<<<NUM_GRAPHS / 16, 256, 0, stream>>>(
        pooled, nullptr, nullptr, nullptr, Wfc, bfc, (float*)d_out, nullptr, nullptr);
}